// TextSAGE_10075993276649
// MI455X (gfx1250) — compile-verified
//
#include <hip/hip_runtime.h>
#include <hip/hip_bf16.h>

// GraphSAGE 2-layer forward for MI455X (gfx1250, wave32).
// N0=900000, N1=43008, N2=2048, D=64, FANOUT=20.

#define D64 64
#define KDIM 128          // concat(self, aggr) = 2*D
#define N1C 43008
#define N2C 2048
#define LDS_STRIDE 132    // 128 + 4 pad -> conflict-free column reads

typedef __attribute__((ext_vector_type(2))) float v2f;
typedef __attribute__((ext_vector_type(8))) float v8f;

// ---------------------------------------------------------------------------
// Scatter-add: 64 threads per edge; wave = 32 contiguous floats (coalesced).
// sums[dst][d] += feat[src][d];  cnts[dst] += 1 (once per edge).
// ---------------------------------------------------------------------------
__global__ __launch_bounds__(256)
void sage_scatter(const float* __restrict__ feat,
                  const int* __restrict__ src,
                  const int* __restrict__ dst,
                  float* __restrict__ sums,
                  float* __restrict__ cnts,
                  int nedges) {
    long long tid = (long long)blockIdx.x * blockDim.x + threadIdx.x;
    int e = (int)(tid >> 6);
    int d = (int)(tid & 63);
    if (e >= nedges) return;
    int s = src[e];
    int t = dst[e];
    atomicAdd(&sums[(long long)t * D64 + d], feat[(long long)s * D64 + d]);
    if (d == 0) atomicAdd(&cnts[t], 1.0f);
}

// ---------------------------------------------------------------------------
// Fused concat + GEMM + bias (+ReLU):
//   out[i,:] = relu?( [self[i,:64] , sums[i,:]/max(cnt[i],1)] @ W[128,64] + b )
// Block = 256 threads = 8 waves; block tile = 32 rows x 64 cols.
// Each wave owns one 16x16 tile, K-loop of 32 x v_wmma_f32_16x16x4_f32.
// Requires n_tgt % 32 == 0 (43008 and 2048 both are) so EXEC is all-1s.
// ---------------------------------------------------------------------------
__global__ __launch_bounds__(256)
void sage_gemm(const float* __restrict__ self_feat,  // [>=n_tgt, 64]
               const float* __restrict__ sums,       // [n_tgt, 64]
               const float* __restrict__ cnts,       // [n_tgt]
               const float* __restrict__ w,          // [128, 64] row-major (k,n)
               const float* __restrict__ b,          // [64]
               float* __restrict__ out,              // [n_tgt, 64]
               int n_tgt, int relu) {
    __shared__ float tile[32 * LDS_STRIDE];

    const int row0 = blockIdx.x * 32;
    const int t = threadIdx.x;

    // Stage 32x128 concat tile into LDS (coalesced: consecutive t -> consecutive k).
    #pragma unroll
    for (int idx = t; idx < 32 * KDIM; idx += 256) {
        int r = idx >> 7;          // 0..31
        int k = idx & 127;         // 0..127
        int row = row0 + r;
        float v;
        if (k < D64) {
            v = self_feat[(long long)row * D64 + k];
        } else {
            float c = cnts[row];
            v = sums[(long long)row * D64 + (k - D64)] / fmaxf(c, 1.0f);
        }
        tile[r * LDS_STRIDE + k] = v;
    }
    __syncthreads();

    const int wave  = t >> 5;       // 0..7
    const int lane  = t & 31;
    const int mtile = wave >> 2;    // 0..1  (row tile within block)
    const int ntile = wave & 3;     // 0..3  (col tile: 4*16 = 64 cols)
    const int lhalf = lane >> 4;    // 0/1 -> K-pair selector
    const int lmod  = lane & 15;

    // A fragment row (ISA 16x4 f32 layout: lanes 0-15 M=0..15 K={0,1}; 16-31 K={2,3})
    const int arow = mtile * 16 + lmod;
    const int ncol = ntile * 16 + lmod;

    v8f acc = {};
    #pragma unroll
    for (int k0 = 0; k0 < KDIM; k0 += 4) {
        int kk = k0 + 2 * lhalf;
        v2f a, bb;
        a.x  = tile[arow * LDS_STRIDE + kk];
        a.y  = tile[arow * LDS_STRIDE + kk + 1];
        bb.x = w[kk * D64 + ncol];          // B 4x16: lanes 0-15 N, K={0,1}/{2,3}
        bb.y = w[(kk + 1) * D64 + ncol];
        acc = __builtin_amdgcn_wmma_f32_16x16x4_f32(
            /*neg_a=*/false, a, /*neg_b=*/false, bb,
            /*c_mod=*/(short)0, acc, /*reuse_a=*/false, /*reuse_b=*/false);
    }

    // C/D layout: lane group selects N; VGPR r -> row (mtile*16 + lhalf*8 + r).
    const float bias = b[ncol];
    const int mbase = row0 + mtile * 16 + lhalf * 8;
    #pragma unroll
    for (int r = 0; r < 8; ++r) {
        float v = acc[r] + bias;
        if (relu) v = fmaxf(v, 0.0f);
        out[(long long)(mbase + r) * D64 + ncol] = v;
    }
}

// ---------------------------------------------------------------------------
// Launch: ws layout (256B aligned):
//   [sums0 | cnt0]  zeroed together, [h], [sums1 | cnt1] zeroed together.
// ---------------------------------------------------------------------------
static inline size_t align256(size_t x) { return (x + 255) & ~(size_t)255; }

extern "C" void kernel_launch(void* const* d_in, const int* in_sizes, int n_in,
                              void* d_out, int out_size, void* d_ws, size_t ws_size,
                              hipStream_t stream) {
    const float* x  = (const float*)d_in[0];
    const float* w0 = (const float*)d_in[1];
    const float* b0 = (const float*)d_in[2];
    const float* w1 = (const float*)d_in[3];
    const float* b1 = (const float*)d_in[4];
    const int* es0  = (const int*)d_in[5];
    const int* ed0  = (const int*)d_in[6];
    const int* es1  = (const int*)d_in[7];
    const int* ed1  = (const int*)d_in[8];
    const int E0 = in_sizes[5];
    const int E1 = in_sizes[7];

    char* ws = (char*)d_ws;
    size_t off = 0;
    const size_t sums0_b = (size_t)N1C * D64 * sizeof(float);
    const size_t cnt0_b  = (size_t)N1C * sizeof(float);
    const size_t h_b     = (size_t)N1C * D64 * sizeof(float);
    const size_t sums1_b = (size_t)N2C * D64 * sizeof(float);
    const size_t cnt1_b  = (size_t)N2C * sizeof(float);

    float* sums0 = (float*)(ws + off);                    // sums0 + cnt0 contiguous
    float* cnt0  = (float*)(ws + off + sums0_b);
    off = align256(off + sums0_b + cnt0_b);
    float* h     = (float*)(ws + off);
    off = align256(off + h_b);
    float* sums1 = (float*)(ws + off);                    // sums1 + cnt1 contiguous
    float* cnt1  = (float*)(ws + off + sums1_b);
    off = align256(off + sums1_b + cnt1_b);
    (void)ws_size; (void)n_in; (void)out_size;

    // ---- Layer 0 ----
    hipMemsetAsync(sums0, 0, sums0_b + cnt0_b, stream);
    {
        long long threads = (long long)E0 * 64;
        int blocks = (int)((threads + 255) / 256);
        sage_scatter<<<blocks, 256, 0, stream>>>(x, es0, ed0, sums0, cnt0, E0);
    }
    sage_gemm<<<N1C / 32, 256, 0, stream>>>(x, sums0, cnt0, w0, b0, h, N1C, 1);

    // ---- Layer 1 ----
    hipMemsetAsync(sums1, 0, sums1_b + cnt1_b, stream);
    {
        long long threads = (long long)E1 * 64;
        int blocks = (int)((threads + 255) / 256);
        sage_scatter<<<blocks, 256, 0, stream>>>(h, es1, ed1, sums1, cnt1, E1);
    }
    sage_gemm<<<N2C / 32, 256, 0, stream>>>(h, sums1, cnt1, w1, b1,
                                            (float*)d_out, N2C, 0);
}